// APPNP_WGTL_77068893159662
// MI455X (gfx1250) — compile-verified
//
#include <hip/hip_runtime.h>
#include <math.h>

#define NFEAT  512
#define NHID   64
#define NCLASS 16
#define KSTEPS 10
#define ALPHA  0.1f
#define WSTRIDE 72   // padded LDS row stride (floats) -> conflict-free B reads

typedef float v2f __attribute__((ext_vector_type(2)));
typedef float v8f __attribute__((ext_vector_type(8)));
typedef int   v4i __attribute__((vector_size(16)));
typedef __attribute__((address_space(1))) v4i as1_v4i;
typedef __attribute__((address_space(3))) v4i as3_v4i;

#if defined(__has_builtin)
#if __has_builtin(__builtin_amdgcn_global_load_async_to_lds_b128) && \
    __has_builtin(__builtin_amdgcn_s_wait_asynccnt)
#define HAS_ASYNC_LDS 1
#endif
#endif

// ---------------- degree / normalization ----------------

__global__ void k_deg_init(float* __restrict__ deg, int n) {
  int i = blockIdx.x * blockDim.x + threadIdx.x;
  if (i < n) deg[i] = 1.0f;  // self-loop contribution
}

__global__ void k_deg_accum(const int* __restrict__ dst, float* __restrict__ deg, int e) {
  int i = blockIdx.x * blockDim.x + threadIdx.x;
  if (i < e) atomicAdd(&deg[dst[i]], 1.0f);
}

__global__ void k_dinv(float* __restrict__ deg, int n) {
  int i = blockIdx.x * blockDim.x + threadIdx.x;
  if (i < n) deg[i] = rsqrtf(deg[i]);  // deg >= 1 always (self loop)
}

__global__ void k_norm(const int* __restrict__ src, const int* __restrict__ dst,
                       const float* __restrict__ dinv, float* __restrict__ norm, int e) {
  int i = blockIdx.x * blockDim.x + threadIdx.x;
  if (i < e) norm[i] = dinv[src[i]] * dinv[dst[i]];
}

// ---------------- h0 = relu(x @ W1 + b1) via fp32 WMMA ----------------
// One wave computes a 16-row x 64-col output tile (4 accumulators).
// W1 (128KB) is staged once per block into LDS (async global->LDS when the
// builtin exists), with padded row stride so the four 16-wide B-fragment
// reads per K-step are LDS bank-conflict free.
// A layout (f32 16x4): lanes 0-15 -> M=lane, K={0,1}; lanes 16-31 -> K={2,3}.
// B layout (f32 4x16): mirrored. C/D: VGPR r -> M=r / M=r+8.
__global__ void __launch_bounds__(256)
k_gemm1_wmma(const float* __restrict__ x, const float* __restrict__ W1,
             const float* __restrict__ b1, float* __restrict__ h0, int n) {
  extern __shared__ float sW1[];  // NFEAT * WSTRIDE floats (147456 B)

  // ---- stage W1 into LDS: 8192 16B chunks, 32 per thread ----
#pragma unroll 4
  for (int it = 0; it < (NFEAT * NHID / 4) / 256; ++it) {
    int cidx = it * 256 + threadIdx.x;       // chunk index
    int row  = cidx >> 4;                    // 0..511
    int c4   = (cidx & 15) << 2;             // 0..60
#ifdef HAS_ASYNC_LDS
    __builtin_amdgcn_global_load_async_to_lds_b128(
        (as1_v4i*)(W1 + (size_t)row * NHID + c4),
        (as3_v4i*)(sW1 + row * WSTRIDE + c4), 0, 0);
#else
    *(float4*)(sW1 + row * WSTRIDE + c4) =
        *(const float4*)(W1 + (size_t)row * NHID + c4);
#endif
  }
#ifdef HAS_ASYNC_LDS
  __builtin_amdgcn_s_wait_asynccnt(0);
#endif
  __syncthreads();

  const int lane = threadIdx.x & 31;
  const int wave = blockIdx.x * (blockDim.x >> 5) + (threadIdx.x >> 5);
  const int row0 = wave * 16;
  if (row0 >= n) return;

  const int m  = lane & 15;
  const int kb = (lane >> 4) << 1;  // 0 or 2 (K half per lane group)
  int rowA = row0 + m;
  if (rowA >= n) rowA = n - 1;      // tail clamp (N=50000 divisible by 16)
  const float* __restrict__ xrow = x + (size_t)rowA * NFEAT;
  const int nc = lane & 15;

  v8f acc0 = {}; v8f acc1 = {}; v8f acc2 = {}; v8f acc3 = {};

  for (int k = 0; k < NFEAT; k += 4) {
    v2f a;
    a.x = xrow[k + kb];
    a.y = xrow[k + kb + 1];
    const float* w0 = sW1 + (k + kb) * WSTRIDE;      // row k+kb
    const float* w1 = w0 + WSTRIDE;                  // row k+kb+1
    v2f b0; b0.x = w0[nc];      b0.y = w1[nc];
    v2f bB; bB.x = w0[16 + nc]; bB.y = w1[16 + nc];
    v2f bC; bC.x = w0[32 + nc]; bC.y = w1[32 + nc];
    v2f bD; bD.x = w0[48 + nc]; bD.y = w1[48 + nc];
    acc0 = __builtin_amdgcn_wmma_f32_16x16x4_f32(false, a, false, b0, (short)0, acc0, false, false);
    acc1 = __builtin_amdgcn_wmma_f32_16x16x4_f32(false, a, false, bB, (short)0, acc1, false, false);
    acc2 = __builtin_amdgcn_wmma_f32_16x16x4_f32(false, a, false, bC, (short)0, acc2, false, false);
    acc3 = __builtin_amdgcn_wmma_f32_16x16x4_f32(false, a, false, bD, (short)0, acc3, false, false);
  }

  const int rbase = (lane >> 4) << 3;  // 0 or 8
#pragma unroll
  for (int t = 0; t < 4; ++t) {
    v8f acc = (t == 0) ? acc0 : (t == 1) ? acc1 : (t == 2) ? acc2 : acc3;
    const float bias = b1[t * 16 + nc];
#pragma unroll
    for (int r = 0; r < 8; ++r) {
      const int row = row0 + rbase + r;
      if (row < n) {
        float v = acc[r] + bias;
        h0[(size_t)row * NHID + t * 16 + nc] = v > 0.f ? v : 0.f;
      }
    }
  }
}

// ---------------- APPNP propagation ----------------
// out = ALPHA*h0 + (1-ALPHA) * (dinv[i]^2 * h[i]  +  sum_edges norm*h[src])

__global__ void k_prop_init(const float* __restrict__ h, const float* __restrict__ h0,
                            const float* __restrict__ dinv, float* __restrict__ out, int n) {
  int t = blockIdx.x * blockDim.x + threadIdx.x;  // n*16 float4 lanes
  if (t >= n * (NHID / 4)) return;
  int i = t >> 4;
  float di = dinv[i];
  float c = (1.0f - ALPHA) * di * di;
  float4 hv  = ((const float4*)h)[t];
  float4 h0v = ((const float4*)h0)[t];
  float4 o;
  o.x = ALPHA * h0v.x + c * hv.x;
  o.y = ALPHA * h0v.y + c * hv.y;
  o.z = ALPHA * h0v.z + c * hv.z;
  o.w = ALPHA * h0v.w + c * hv.w;
  ((float4*)out)[t] = o;
}

__global__ void k_prop_edges(const int* __restrict__ src, const int* __restrict__ dst,
                             const float* __restrict__ norm, const float* __restrict__ h,
                             float* __restrict__ out, int e) {
  int t = blockIdx.x * blockDim.x + threadIdx.x;  // e*16 threads, float4 each
  if (t >= e * (NHID / 4)) return;
  int ei = t >> 4;
  int f4 = (t & 15) << 2;
  float w = (1.0f - ALPHA) * norm[ei];
  float4 hv = *(const float4*)(h + (size_t)src[ei] * NHID + f4);
  float* o = out + (size_t)dst[ei] * NHID + f4;
  atomicAdd(o + 0, w * hv.x);
  atomicAdd(o + 1, w * hv.y);
  atomicAdd(o + 2, w * hv.z);
  atomicAdd(o + 3, w * hv.w);
}

// ---------------- attention aggregation + W2 transform ----------------
__global__ void __launch_bounds__(256)
k_attn(const float* __restrict__ h, const float* __restrict__ lt,
       const float* __restrict__ gt,
       const float* __restrict__ Wa1, const float* __restrict__ ba1,
       const float* __restrict__ Wa2, const float* __restrict__ W2,
       float* __restrict__ y, int n) {
  __shared__ float sagg[8][NHID];
  const int lane = threadIdx.x & 31;
  const int w    = threadIdx.x >> 5;
  const int node = blockIdx.x * 8 + w;
  const bool active = node < n;
  const int nd = active ? node : 0;
  const float* __restrict__ hrow = h  + (size_t)nd * NHID;
  const float* __restrict__ lrow = lt + (size_t)nd * NHID;

  float a0 = ba1[lane], a1 = ba1[lane];
  for (int k = 0; k < NHID; ++k) {
    float wv = Wa1[k * 32 + lane];
    a0 = fmaf(hrow[k], wv, a0);
    a1 = fmaf(lrow[k], wv, a1);
  }
  float s = Wa2[lane];
  float t0 = tanhf(a0) * s;
  float t1 = tanhf(a1) * s;
  for (int off = 16; off; off >>= 1) {
    t0 += __shfl_xor(t0, off, 32);
    t1 += __shfl_xor(t1, off, 32);
  }
  float mx = fmaxf(t0, t1);
  float e0 = expf(t0 - mx), e1 = expf(t1 - mx);
  float inv = 1.0f / (e0 + e1);
  float be0 = e0 * inv, be1 = e1 * inv;

  float g0 = (be0 * hrow[lane]      + be1 * lrow[lane])      * gt[lane];
  float g1 = (be0 * hrow[lane + 32] + be1 * lrow[lane + 32]) * gt[lane + 32];
  sagg[w][lane]      = g0;
  sagg[w][lane + 32] = g1;
  __syncthreads();

  const int c = lane & 15, half = lane >> 4;
  float acc = 0.0f;
  for (int f = half * 32; f < half * 32 + 32; ++f)
    acc = fmaf(sagg[w][f], W2[f * NCLASS + c], acc);
  acc += __shfl_xor(acc, 16, 32);
  if (active && half == 0) y[(size_t)node * NCLASS + c] = acc;
}

// ---------------- final GCN prop + log_softmax ----------------

__global__ void k_final_init(const float* __restrict__ y, const float* __restrict__ dinv,
                             const float* __restrict__ b2, float* __restrict__ out, int n) {
  int t = blockIdx.x * blockDim.x + threadIdx.x;  // n*4 float4 lanes
  if (t >= n * (NCLASS / 4)) return;
  int i = t >> 2;
  float di = dinv[i];
  float c = di * di;
  float4 yv = ((const float4*)y)[t];
  float4 bv = ((const float4*)b2)[t & 3];
  float4 o;
  o.x = c * yv.x + bv.x;
  o.y = c * yv.y + bv.y;
  o.z = c * yv.z + bv.z;
  o.w = c * yv.w + bv.w;
  ((float4*)out)[t] = o;
}

__global__ void k_final_edges(const int* __restrict__ src, const int* __restrict__ dst,
                              const float* __restrict__ norm, const float* __restrict__ y,
                              float* __restrict__ out, int e) {
  int t = blockIdx.x * blockDim.x + threadIdx.x;  // e*4 threads, float4 each
  if (t >= e * (NCLASS / 4)) return;
  int ei = t >> 2;
  int c4 = (t & 3) << 2;
  float w = norm[ei];
  float4 yv = *(const float4*)(y + (size_t)src[ei] * NCLASS + c4);
  float* o = out + (size_t)dst[ei] * NCLASS + c4;
  atomicAdd(o + 0, w * yv.x);
  atomicAdd(o + 1, w * yv.y);
  atomicAdd(o + 2, w * yv.z);
  atomicAdd(o + 3, w * yv.w);
}

__global__ void k_logsoftmax(float* __restrict__ out, int n) {
  int i = blockIdx.x * blockDim.x + threadIdx.x;
  if (i >= n) return;
  float4 v[4];
#pragma unroll
  for (int q = 0; q < 4; ++q) v[q] = ((const float4*)out)[i * 4 + q];
  float mx = -INFINITY;
#pragma unroll
  for (int q = 0; q < 4; ++q) {
    mx = fmaxf(mx, fmaxf(fmaxf(v[q].x, v[q].y), fmaxf(v[q].z, v[q].w)));
  }
  float ssum = 0.0f;
#pragma unroll
  for (int q = 0; q < 4; ++q) {
    ssum += expf(v[q].x - mx) + expf(v[q].y - mx) + expf(v[q].z - mx) + expf(v[q].w - mx);
  }
  float lse = mx + logf(ssum);
#pragma unroll
  for (int q = 0; q < 4; ++q) {
    v[q].x -= lse; v[q].y -= lse; v[q].z -= lse; v[q].w -= lse;
    ((float4*)out)[i * 4 + q] = v[q];
  }
}

// ---------------- host launch ----------------

extern "C" void kernel_launch(void* const* d_in, const int* in_sizes, int n_in,
                              void* d_out, int out_size, void* d_ws, size_t ws_size,
                              hipStream_t stream) {
  const float* x   = (const float*)d_in[0];
  const int*   ei  = (const int*)  d_in[1];
  const float* lt  = (const float*)d_in[2];
  const float* gt  = (const float*)d_in[3];
  const float* W1  = (const float*)d_in[4];
  const float* b1  = (const float*)d_in[5];
  const float* Wa1 = (const float*)d_in[6];
  const float* ba1 = (const float*)d_in[7];
  const float* Wa2 = (const float*)d_in[8];
  const float* W2  = (const float*)d_in[9];
  const float* b2  = (const float*)d_in[10];

  const int n = in_sizes[0] / NFEAT;
  const int e = in_sizes[1] / 2;
  const int* src = ei;
  const int* dst = ei + e;
  float* out = (float*)d_out;

  // workspace layout (floats)
  float* ws   = (float*)d_ws;
  float* dinv = ws;                       // n
  float* norm = dinv + n;                 // e
  float* h0   = norm + e;                 // n*64
  float* ha   = h0 + (size_t)n * NHID;    // n*64
  float* hb   = ha + (size_t)n * NHID;    // n*64
  float* yv   = hb + (size_t)n * NHID;    // n*16

  const int T = 256;
  // degrees -> dinv -> per-edge norm
  k_deg_init <<<(n + T - 1) / T, T, 0, stream>>>(dinv, n);
  k_deg_accum<<<(e + T - 1) / T, T, 0, stream>>>(dst, dinv, e);
  k_dinv     <<<(n + T - 1) / T, T, 0, stream>>>(dinv, n);
  k_norm     <<<(e + T - 1) / T, T, 0, stream>>>(src, dst, dinv, norm, e);

  // h0 = relu(x @ W1 + b1) : one wave per 16 rows, W1 staged in LDS
  {
    int nwaves = (n + 15) / 16;
    int blocks = (nwaves + 7) / 8;  // 8 waves per 256-thread block
    size_t lds = (size_t)NFEAT * WSTRIDE * sizeof(float);  // 147456 B
    k_gemm1_wmma<<<blocks, T, lds, stream>>>(x, W1, b1, h0, n);
  }

  // K-step APPNP propagation (ping-pong)
  const float* cur = h0;
  float* bufs[2] = {ha, hb};
  for (int s = 0; s < KSTEPS; ++s) {
    float* nxt = bufs[s & 1];
    k_prop_init <<<(n * (NHID / 4) + T - 1) / T, T, 0, stream>>>(cur, h0, dinv, nxt, n);
    k_prop_edges<<<(e * (NHID / 4) + T - 1) / T, T, 0, stream>>>(src, dst, norm, cur, nxt, e);
    cur = nxt;
  }

  // attention aggregation + W2 transform
  k_attn<<<(n + 7) / 8, T, 0, stream>>>(cur, lt, gt, Wa1, ba1, Wa2, W2, yv, n);

  // final GCN propagation into d_out, then in-place log_softmax
  k_final_init <<<(n * (NCLASS / 4) + T - 1) / T, T, 0, stream>>>(yv, dinv, b2, out, n);
  k_final_edges<<<(e * (NCLASS / 4) + T - 1) / T, T, 0, stream>>>(src, dst, norm, yv, out, e);
  k_logsoftmax <<<(n + T - 1) / T, T, 0, stream>>>(out, n);
}